// ReasoningMamba2Block_18356690223122
// MI455X (gfx1250) — compile-verified
//
#include <hip/hip_runtime.h>
#include <hip/hip_bf16.h>

// Problem constants (from reference)
#define Bn   64
#define En   2048
#define DIn  4096
#define Sn   128
#define DTRn 128
#define Rn   16

typedef __attribute__((ext_vector_type(2))) float v2f;
typedef __attribute__((ext_vector_type(4))) float v4f;
typedef __attribute__((ext_vector_type(8))) float v8f;

// ---------------------------------------------------------------------------
// Weight repacks (tiny, run every call; deterministic)
// Bg2[j*(R*o2) + r*o2 + q] = g2[r,j,q]
__global__ void pack_g2_k(const float* __restrict__ g2, float* __restrict__ out,
                          int R, int i2, int o2) {
  int idx = blockIdx.x * blockDim.x + threadIdx.x;
  int total = R * i2 * o2;
  if (idx >= total) return;
  int q = idx % o2;
  int j = (idx / o2) % i2;
  int r = idx / (o2 * i2);
  out[j * (R * o2) + r * o2 + q] = g2[idx];
}

// G1p[p*(i1*R) + i*R + r] = g1[i,p,r]
__global__ void pack_g1_k(const float* __restrict__ g1, float* __restrict__ out,
                          int i1, int o1, int R) {
  int idx = blockIdx.x * blockDim.x + threadIdx.x;
  int total = i1 * o1 * R;
  if (idx >= total) return;
  int r = idx % R;
  int p = (idx / R) % o1;
  int i = idx / (R * o1);
  out[p * (i1 * R) + i * R + r] = g1[idx];
}

// ---------------------------------------------------------------------------
// Generic batched fp32 GEMM on V_WMMA_F32_16X16X4_F32.
// One wave computes a 16x32 C tile (two 16x16 WMMA tiles sharing the A
// operand). K must be a multiple of 4. OOB lanes are *clamped* to valid
// addresses (their products only reach C rows>=M / cols>=N, masked at the
// store) -> the inner loop has NO divergent branches and pure pointer-bump
// addressing.
__global__ void gemm_wmma_f32(const float* __restrict__ A,
                              const float* __restrict__ Bm,
                              float* __restrict__ C,
                              const float* __restrict__ bias,
                              int M, int N, int K,
                              int lda, int ldb, int ldc,
                              long sA, long sB, long sC,
                              int nWaves, int wavesN) {
  int waveId = blockIdx.x * (blockDim.x >> 5) + (threadIdx.x >> 5);
  if (waveId >= nWaves) return;               // wave-uniform: EXEC stays full
  int lane = threadIdx.x & 31;
  int wm = waveId / wavesN;
  int wn = waveId % wavesN;
  A  += (long)blockIdx.z * sA;
  Bm += (long)blockIdx.z * sB;
  C  += (long)blockIdx.z * sC;

  int row0 = wm << 4;
  int col0 = wn << 5;                         // 32 columns per wave
  int lhalf = lane >> 4;                      // 0: K=0,1 | 1: K=2,3
  int l15   = lane & 15;
  int koff  = lhalf << 1;

  int ar  = row0 + l15;                       // A row for this lane
  int bc0 = col0 + l15;                       // B/C column, tile 0
  int bc1 = bc0 + 16;                         // B/C column, tile 1
  int arc  = ar  < M ? ar  : 0;               // clamp: keep addresses legal
  int bc0c = bc0 < N ? bc0 : 0;
  int bc1c = bc1 < N ? bc1 : 0;

  const float* ap  = A  + (long)arc * lda + koff;     // K,K+1 contiguous
  const float* bp0 = Bm + (long)koff * ldb + bc0c;
  const float* bp1 = Bm + (long)koff * ldb + bc1c;
  const long bstep = (long)ldb * 4;

  v8f acc0 = {}, acc1 = {};
  for (int k = 0; k < K; k += 4) {
    v2f a = *(const v2f*)ap;
    v2f b0, b1;
    b0.x = bp0[0]; b0.y = bp0[ldb];
    b1.x = bp1[0]; b1.y = bp1[ldb];
    // (neg_a, A, neg_b, B, c_mod, C, reuse_a, reuse_b)
    acc0 = __builtin_amdgcn_wmma_f32_16x16x4_f32(false, a, false, b0,
                                                 (short)0, acc0, false, false);
    acc1 = __builtin_amdgcn_wmma_f32_16x16x4_f32(false, a, false, b1,
                                                 (short)0, acc1, false, false);
    ap  += 4;
    bp0 += bstep;
    bp1 += bstep;
  }

  int rbase = row0 + (lhalf << 3);            // acc[v] -> M = rbase + v
#pragma unroll
  for (int v = 0; v < 8; ++v) {
    int r = rbase + v;
    if (r < M) {
      if (bc0 < N) {
        float val = acc0[v];
        if (bias) val += bias[r * ldc + bc0];
        C[(long)r * ldc + bc0] = val;
      }
      if (bc1 < N) {
        float val = acc1[v];
        if (bias) val += bias[r * ldc + bc1];
        C[(long)r * ldc + bc1] = val;
      }
    }
  }
}

// ---------------------------------------------------------------------------
// Depthwise conv tail (+ SiLU) and new_conv_state.
// SAME padding, window 4, length 4, last position -> uses pad idx 2..5:
//   conv_out = cs2*f0 + x_c*f1 + bias
__global__ void conv_silu_k(const float* __restrict__ xz,
                            const float* __restrict__ conv_state,
                            const float* __restrict__ filt,
                            const float* __restrict__ cbias,
                            float* __restrict__ xa,
                            float* __restrict__ new_cs) {
  int idx = blockIdx.x * blockDim.x + threadIdx.x;   // b*DI + c
  if (idx >= Bn * DIn) return;
  int b = idx >> 12;
  int c = idx & (DIn - 1);
  float xc  = xz[(b << 13) + c];                     // x_c = xz[b][0:4096]
  float cs1 = conv_state[((b * 3) + 1) * DIn + c];
  float cs2 = conv_state[((b * 3) + 2) * DIn + c];
  float co  = cs2 * filt[c] + xc * filt[DIn + c] + cbias[c];
  xa[idx] = co / (1.f + __expf(-co));                // silu
  new_cs[(b * 3 + 0) * DIn + c] = cs1;
  new_cs[(b * 3 + 1) * DIn + c] = cs2;
  new_cs[(b * 3 + 2) * DIn + c] = xc;
}

// ---------------------------------------------------------------------------
// Fused SSM update (bandwidth-bound core): one wave32 per (b,c) row.
// Each lane owns 4 contiguous states (128b), full 512B coalesced per wave.
// h_padded / new_h are 2x134MB of pure streaming traffic: non-temporal so
// they do not evict the reused operands (A_log 64x reuse, dtbc 4096x reuse)
// from the 192MB L2.
__global__ void ssm_kernel(const float* __restrict__ h_padded,
                           const float* __restrict__ xa,
                           const float* __restrict__ dtbc,
                           const float* __restrict__ dt_raw,
                           const float* __restrict__ xz,
                           const float* __restrict__ A_log,
                           const float* __restrict__ Dv,
                           float* __restrict__ new_h,
                           float* __restrict__ ygate) {
  int wave = (blockIdx.x * blockDim.x + threadIdx.x) >> 5;
  if (wave >= Bn * DIn) return;
  int lane = threadIdx.x & 31;
  int b = wave >> 12;
  int c = wave & (DIn - 1);

  float dtv = dt_raw[wave];
  dtv = (dtv > 20.f) ? dtv : log1pf(__expf(dtv));    // softplus
  float xav  = xa[wave];
  float dtxa = dtv * xav;

  const v4f* h4p = (const v4f*)(h_padded + ((size_t)wave << 7));
  v4f*       n4p = (v4f*)(new_h + ((size_t)wave << 7));
  const v4f* a4p = (const v4f*)(A_log + ((size_t)c << 7));
  const v4f* bm4 = (const v4f*)(dtbc + b * 384 + DTRn);
  const v4f* cm4 = (const v4f*)(dtbc + b * 384 + DTRn + Sn);

  v4f h4 = __builtin_nontemporal_load(h4p + lane);
  v4f a4 = a4p[lane];
  v4f b4 = bm4[lane];
  v4f c4 = cm4[lane];
  v4f o4;
  o4.x = __expf(dtv * -__expf(a4.x)) * h4.x + dtxa * b4.x;
  o4.y = __expf(dtv * -__expf(a4.y)) * h4.y + dtxa * b4.y;
  o4.z = __expf(dtv * -__expf(a4.z)) * h4.z + dtxa * b4.z;
  o4.w = __expf(dtv * -__expf(a4.w)) * h4.w + dtxa * b4.w;
  __builtin_nontemporal_store(o4, n4p + lane);

  float acc = o4.x * c4.x + o4.y * c4.y + o4.z * c4.z + o4.w * c4.w;
#pragma unroll
  for (int off = 16; off > 0; off >>= 1) acc += __shfl_down(acc, off, 32);

  if (lane == 0) {
    float y  = acc + xav * Dv[c];
    float zv = xz[(b << 13) + DIn + c];
    ygate[wave] = y * (zv / (1.f + __expf(-zv)));
  }
}

// ---------------------------------------------------------------------------
extern "C" void kernel_launch(void* const* d_in, const int* in_sizes, int n_in,
                              void* d_out, int out_size, void* d_ws, size_t ws_size,
                              hipStream_t stream) {
  const float* x          = (const float*)d_in[0];
  const float* h_padded   = (const float*)d_in[1];
  const float* conv_state = (const float*)d_in[2];
  const float* in_g1 = (const float*)d_in[3];
  const float* in_g2 = (const float*)d_in[4];
  const float* in_b  = (const float*)d_in[5];
  const float* xp_g1 = (const float*)d_in[6];
  const float* xp_g2 = (const float*)d_in[7];
  const float* xp_b  = (const float*)d_in[8];
  const float* dt_g1 = (const float*)d_in[9];
  const float* dt_g2 = (const float*)d_in[10];
  const float* dt_b  = (const float*)d_in[11];
  const float* op_g1 = (const float*)d_in[12];
  const float* op_g2 = (const float*)d_in[13];
  const float* op_b  = (const float*)d_in[14];
  const float* conv_filter = (const float*)d_in[15];
  const float* conv_bias   = (const float*)d_in[16];
  const float* A_log = (const float*)d_in[17];
  const float* Dv    = (const float*)d_in[18];

  float* out    = (float*)d_out;                      // 64*2048
  float* new_h  = out + Bn * En;                      // 64*4096*128
  float* new_cs = new_h + (size_t)Bn * DIn * Sn;      // 64*3*4096

  float* ws     = (float*)d_ws;
  float* xz     = ws;                 // 524288  (x_c | z)
  float* xa     = xz + 524288;        // 262144
  float* dtbc   = xa + 262144;        // 24576   (dt_u | Bm | Cm)
  float* dt_raw = dtbc + 24576;       // 262144
  float* ygate  = dt_raw + 262144;    // 262144
  float* Bg2_in = ygate + 262144;     // 131072
  float* Bg2_xp = Bg2_in + 131072;    // 24576
  float* Bg2_dt = Bg2_xp + 24576;     // 16384
  float* Bg2_op = Bg2_dt + 16384;     // 65536
  float* G1in   = Bg2_op + 65536;     // 32768
  float* G1xp   = G1in + 32768;       // 16384
  float* G1dt   = G1xp + 16384;       // 8192
  float* G1op   = G1dt + 8192;        // 32768
  float* u      = G1op + 32768;       // up to 4194304 (reused across layers)

  auto cdiv = [](int a, int b) { return (a + b - 1) / b; };
  auto gemm = [&](const float* A, const float* Bm, float* C, const float* bias,
                  int M, int N, int K, int lda, int ldb, int ldc,
                  long sA, long sB, long sC, int batch) {
    int wavesM = cdiv(M, 16), wavesN = cdiv(N, 32);
    int nW = wavesM * wavesN;
    dim3 grid(cdiv(nW, 4), 1, batch);
    gemm_wmma_f32<<<grid, 128, 0, stream>>>(A, Bm, C, bias, M, N, K,
                                            lda, ldb, ldc, sA, sB, sC, nW, wavesN);
  };

  // ---- pack TT cores ----
  pack_g2_k<<<cdiv(16*64*128,256),256,0,stream>>>(in_g2, Bg2_in, 16, 64, 128);
  pack_g2_k<<<cdiv(16*64*24 ,256),256,0,stream>>>(xp_g2, Bg2_xp, 16, 64, 24);
  pack_g2_k<<<cdiv(16*16*64 ,256),256,0,stream>>>(dt_g2, Bg2_dt, 16, 16, 64);
  pack_g2_k<<<cdiv(16*64*64 ,256),256,0,stream>>>(op_g2, Bg2_op, 16, 64, 64);
  pack_g1_k<<<cdiv(32*64*16 ,256),256,0,stream>>>(in_g1, G1in, 32, 64, 16);
  pack_g1_k<<<cdiv(64*16*16 ,256),256,0,stream>>>(xp_g1, G1xp, 64, 16, 16);
  pack_g1_k<<<cdiv(8*64*16  ,256),256,0,stream>>>(dt_g1, G1dt, 8, 64, 16);
  pack_g1_k<<<cdiv(64*32*16 ,256),256,0,stream>>>(op_g1, G1op, 64, 32, 16);

  // ---- in-projection TT: x (64,2048) -> xz (64,8192) ----
  // stage1: (B*i1=2048 x 64) @ (64 x R*o2=2048) -> u
  gemm(x, Bg2_in, u, nullptr, 2048, 2048, 64, 64, 2048, 2048, 0, 0, 0, 1);
  // stage2 (batched b): (64 x 512) @ (512 x 128) -> xz[b]
  gemm(G1in, u, xz, in_b, 64, 128, 512, 512, 128, 128, 0, 65536, 8192, 64);

  // ---- conv tail + silu + new_conv_state ----
  conv_silu_k<<<cdiv(Bn*DIn,256),256,0,stream>>>(xz, conv_state, conv_filter,
                                                 conv_bias, xa, new_cs);

  // ---- x-projection TT: xa (64,4096) -> dtbc (64,384) ----
  gemm(xa, Bg2_xp, u, nullptr, 4096, 384, 64, 64, 384, 384, 0, 0, 0, 1);
  gemm(G1xp, u, dtbc, xp_b, 16, 24, 1024, 1024, 24, 24, 0, 24576, 384, 64);

  // ---- dt TT: dt_u (64,128) -> dt_raw (64,4096) ----
  // stage1 batched over b: A = dtbc[b][0:128] viewed as 8x16 (lda=16, stride 384)
  gemm(dtbc, Bg2_dt, u, nullptr, 8, 1024, 16, 16, 1024, 1024, 384, 0, 8192, 64);
  gemm(G1dt, u, dt_raw, dt_b, 64, 64, 128, 128, 64, 64, 0, 8192, 4096, 64);

  // ---- fused SSM update (HBM-bound: 268 MB of h traffic) ----
  ssm_kernel<<<Bn*DIn/8, 256, 0, stream>>>(h_padded, xa, dtbc, dt_raw, xz,
                                           A_log, Dv, new_h, ygate);

  // ---- out-projection TT: ygate (64,4096) -> out (64,2048) ----
  gemm(ygate, Bg2_op, u, nullptr, 4096, 1024, 64, 64, 1024, 1024, 0, 0, 0, 1);
  gemm(G1op, u, out, op_b, 32, 64, 1024, 1024, 64, 64, 0, 65536, 2048, 64);
}